// SetAbstraction_39324720562663
// MI455X (gfx1250) — compile-verified
//
#include <hip/hip_runtime.h>
#include <math.h>

// ---------------------------------------------------------------------------
// PointNet++ SetAbstraction on gfx1250 (MI455X), wave32, WMMA f32 16x16x4.
// Pipeline: FPS -> gather centers -> ball query (top-32) -> gather local ->
//   GEMM1(WMMA)+stats -> BN fin -> GEMM2(WMMA, BN+GELU fused in A-stage) ->
//   BN fin -> skip GEMM(WMMA) -> maxpool(BN+GELU) -> GEMM3(WMMA)+stats ->
//   BN fin -> GEMM4(WMMA, BN+GELU fused)+stats -> BN fin -> residual add.
// GEMMs: block = 8 waves = 16-row slab x 128-col group; A & W K-chunks staged
// in LDS (bank-conflict-free pair-interleaved layouts); MODE-0 staging uses
// gfx1250 async global->LDS loads (ASYNCcnt); inner loop is pure
// ds_load_b64 + v_wmma. Deterministic BN stats via per-block partials.
// ---------------------------------------------------------------------------

#define BB    8
#define NN    8192
#define MM    2048
#define KK    32
#define CIN   64
#define COUT  128
#define CMID  512
#define LIN   68            // 3 + 64 padded to multiple of 4
#define RLOC  (BB*MM*KK)    // 524288
#define RPTS  (BB*MM)       // 16384
#define R2    0.0225f       // radius^2
#define BIGF  1.0e10f

#define WPAIR_STRIDE 288    // floats per W pair-row: 256 + 32 (bank offset)

typedef float v2f __attribute__((ext_vector_type(2)));
typedef float v8f __attribute__((ext_vector_type(8)));

__device__ __forceinline__ float gelu_f(float x) {
    return 0.5f * x * (1.0f + erff(x * 0.70710678118654752440f));
}

// gfx1250 async global->LDS b32 copy: per-lane LDS destination address in a
// VGPR, per-lane 64-bit global source address. Tracked by ASYNCcnt.
__device__ __forceinline__ void async_ld_b32(unsigned lds_addr, const float* gptr) {
    asm volatile("global_load_async_to_lds_b32 %0, %1, off"
                 :: "v"(lds_addr), "v"(gptr) : "memory");
}
__device__ __forceinline__ void wait_async0() {
    asm volatile("s_wait_asynccnt 0x0" ::: "memory");
}

// ---------------------------------------------------------------------------
// Farthest point sampling: 1 block per batch, 1024 threads, 8 pts/thread in
// registers. Center coords broadcast through LDS (no global load in the
// serial loop). argmax tie-break = lowest index (matches jnp.argmax).
// ---------------------------------------------------------------------------
__global__ __launch_bounds__(1024) void fps_kernel(const float* __restrict__ pos,
                                                   int* __restrict__ cidx) {
    const int b = blockIdx.x;
    const int t = threadIdx.x;
    const float* P = pos + (size_t)b * NN * 3;
    __shared__ float cs[3];
    __shared__ float rv[32];
    __shared__ int   ri[32];
    __shared__ int   far_s;

    float px[8], py[8], pz[8], dist[8];
#pragma unroll
    for (int j = 0; j < 8; ++j) {
        int n = t + j * 1024;
        px[j] = P[n * 3 + 0];
        py[j] = P[n * 3 + 1];
        pz[j] = P[n * 3 + 2];
        dist[j] = BIGF;
    }
    if (t == 0) { far_s = 0; cs[0] = px[0]; cs[1] = py[0]; cs[2] = pz[0]; }
    __syncthreads();
    int far = 0;

    for (int i = 0; i < MM; ++i) {
        if (t == 0) cidx[(size_t)b * MM + i] = far;
        float cx = cs[0], cy = cs[1], cz = cs[2];
        float bv = -1.0f; int bi = 0;
#pragma unroll
        for (int j = 0; j < 8; ++j) {
            float dx = px[j] - cx, dy = py[j] - cy, dz = pz[j] - cz;
            float d = dx * dx + dy * dy + dz * dz;
            d = fminf(dist[j], d);
            dist[j] = d;
            int n = t + j * 1024;
            if (d > bv) { bv = d; bi = n; }
        }
#pragma unroll
        for (int off = 16; off > 0; off >>= 1) {
            float ov = __shfl_down(bv, off, 32);
            int   oi = __shfl_down(bi, off, 32);
            if (ov > bv || (ov == bv && oi < bi)) { bv = ov; bi = oi; }
        }
        if ((t & 31) == 0) { rv[t >> 5] = bv; ri[t >> 5] = bi; }
        __syncthreads();
        if (t < 32) {
            float bv2 = rv[t]; int bi2 = ri[t];
#pragma unroll
            for (int off = 16; off > 0; off >>= 1) {
                float ov = __shfl_down(bv2, off, 32);
                int   oi = __shfl_down(bi2, off, 32);
                if (ov > bv2 || (ov == bv2 && oi < bi2)) { bv2 = ov; bi2 = oi; }
            }
            if (t == 0) far_s = bi2;
        }
        __syncthreads();
        far = far_s;
        if ((far & 1023) == t) {           // owner broadcasts new center coords
            int jw = far >> 10;
#pragma unroll
            for (int j = 0; j < 8; ++j)
                if (j == jw) { cs[0] = px[j]; cs[1] = py[j]; cs[2] = pz[j]; }
        }
        __syncthreads();
    }
}

// ---------------------------------------------------------------------------
__global__ __launch_bounds__(256) void gather_centers(const float* __restrict__ feat,
                                                      const float* __restrict__ pos,
                                                      const int* __restrict__ cidx,
                                                      float* __restrict__ cpos,
                                                      float* __restrict__ cfeat,
                                                      float* __restrict__ cpos_out) {
    const int t  = threadIdx.x;
    const int r  = blockIdx.x * 4 + (t >> 6);
    const int cl = t & 63;
    const int b  = r >> 11;                  // MM = 2048
    const int ci = cidx[r];
    cfeat[(size_t)r * CIN + cl] = feat[((size_t)b * NN + ci) * CIN + cl];
    if (cl < 3) {
        float p = pos[((size_t)b * NN + ci) * 3 + cl];
        cpos[r * 3 + cl]     = p;
        cpos_out[r * 3 + cl] = p;
    }
}

// ---------------------------------------------------------------------------
// Ball query: 1 block per center, 256 threads, 32 pts/thread (d2 in regs).
// d2 = max(0,|c|^2+|p|^2-2c.p); >r^2 -> INF; pick 32 smallest (ties ->
// lowest index); INF slots -> fallback to nearest. 32 block-argmin passes.
// ---------------------------------------------------------------------------
__global__ __launch_bounds__(256) void ballquery_kernel(const float* __restrict__ pos,
                                                        const float* __restrict__ cpos,
                                                        int* __restrict__ knn) {
    const int bm = blockIdx.x;
    const int b  = bm >> 11;
    const int t  = threadIdx.x;
    const float* P = pos + (size_t)b * NN * 3;
    __shared__ float rv[8];
    __shared__ int   ri[8];
    __shared__ float bestv_s;
    __shared__ int   besti_s;

    const float cx = cpos[bm * 3 + 0], cy = cpos[bm * 3 + 1], cz = cpos[bm * 3 + 2];
    const float cc = cx * cx + cy * cy + cz * cz;

    float d2[32];
#pragma unroll
    for (int j = 0; j < 32; ++j) {
        int n = t + j * 256;
        float x = P[n * 3 + 0], y = P[n * 3 + 1], z = P[n * 3 + 2];
        float pp = x * x + y * y + z * z;
        float dt = cx * x + cy * y + cz * z;
        float d = cc + pp - 2.0f * dt;
        d = fmaxf(d, 0.0f);
        if (d > R2) d = BIGF;
        d2[j] = d;
    }

    int first = 0;
    for (int kk = 0; kk < KK; ++kk) {
        float bv = 2.0f * BIGF; int bi = 0x7fffffff;
#pragma unroll
        for (int j = 0; j < 32; ++j) {
            int n = t + j * 256;
            if (d2[j] < bv) { bv = d2[j]; bi = n; }
        }
#pragma unroll
        for (int off = 16; off > 0; off >>= 1) {
            float ov = __shfl_down(bv, off, 32);
            int   oi = __shfl_down(bi, off, 32);
            if (ov < bv || (ov == bv && oi < bi)) { bv = ov; bi = oi; }
        }
        if ((t & 31) == 0) { rv[t >> 5] = bv; ri[t >> 5] = bi; }
        __syncthreads();
        if (t < 32) {
            float bv2 = (t < 8) ? rv[t] : 2.0f * BIGF;
            int   bi2 = (t < 8) ? ri[t] : 0x7fffffff;
#pragma unroll
            for (int off = 16; off > 0; off >>= 1) {
                float ov = __shfl_down(bv2, off, 32);
                int   oi = __shfl_down(bi2, off, 32);
                if (ov < bv2 || (ov == bv2 && oi < bi2)) { bv2 = ov; bi2 = oi; }
            }
            if (t == 0) { bestv_s = bv2; besti_s = bi2; }
        }
        __syncthreads();
        const float bestv = bestv_s;
        const int   besti = besti_s;
        if ((besti & 255) == t) {            // invalidate winner slot
            int jw = besti >> 8;
#pragma unroll
            for (int j = 0; j < 32; ++j)
                if (j == jw) d2[j] = 2.0f * BIGF;
        }
        if (t == 0) {
            if (kk == 0) first = besti;
            knn[(size_t)bm * KK + kk] = (bestv >= BIGF) ? first : besti;
        }
        __syncthreads();
    }
}

// ---------------------------------------------------------------------------
__global__ __launch_bounds__(256) void gather_local(const float* __restrict__ feat,
                                                    const float* __restrict__ pos,
                                                    const int* __restrict__ knn,
                                                    const float* __restrict__ cpos,
                                                    float* __restrict__ li) {
    const int t  = threadIdx.x;
    const int r  = blockIdx.x * 4 + (t >> 6);
    const int cl = t & 63;
    const int bm = r >> 5;
    const int b  = bm >> 11;
    const int n  = knn[r];
    float* row = li + (size_t)r * LIN;
    row[3 + cl] = feat[((size_t)b * NN + n) * CIN + cl];
    if (cl < 3) row[cl] = pos[((size_t)b * NN + n) * 3 + cl] - cpos[bm * 3 + cl];
    if (cl == 3) row[67] = 0.0f;
}

// pad lw1 (67x128) -> (68x128) with zero last row
__global__ void pad_lw1_kernel(const float* __restrict__ lw1, float* __restrict__ lw1p) {
    int i = blockIdx.x * blockDim.x + threadIdx.x;
    if (i >= LIN * COUT) return;
    int r = i >> 7;
    lw1p[i] = (r < 67) ? lw1[i] : 0.0f;
}

// ---------------------------------------------------------------------------
// WMMA GEMM core: Y[R x C] = op(A)[R x Kd] @ W[Kd x C] + bias.
// MODE 0: op = identity, staging via async global->LDS (ASYNCcnt);
// MODE 2: op = GELU(BN(.)) applied in registers during synchronous staging.
// Block (256 thr, 8 waves) owns: 16-row slab (grid-strided via blockIdx.x)
// x 128-column group (blockIdx.y). K processed in chunks of KC (<=128).
// LDS layouts (all v2f accesses bank-conflict-free across the wave halves):
//   Ws[k2][c][2] pair-interleaved, pair stride WPAIR_STRIDE=288 (== 32 mod 64)
//   As[k2][row][2] pair-interleaved, pair stride 32            (== 32 mod 64)
// Inner loop: ds_load_b64(A) + ds_load_b64(B) + v_wmma_f32_16x16x4_f32.
// Deterministic BN stats: per-block column partial sums/sumsq.
// ---------------------------------------------------------------------------
template <int MODE>
__global__ __launch_bounds__(256) void gemm_core(
    const float* __restrict__ A, int lda,
    const float* __restrict__ W,
    const float* __restrict__ bias,
    const float* __restrict__ inS, const float* __restrict__ inB,
    float* __restrict__ Y,
    float* __restrict__ partS, float* __restrict__ partQ,
    int R, int C, int Kd, int KC)
{
    extern __shared__ float smem[];
    float* Ws = smem;                             // (KC/2) * WPAIR_STRIDE floats
    float* As = smem + (KC >> 1) * WPAIR_STRIDE;  // (KC/2) * 32 floats
    // LDS byte addresses for async copies (dynamic LDS starts after static)
    const unsigned ldsWs = __builtin_amdgcn_groupstaticsize();
    const unsigned ldsAs = ldsWs + (unsigned)((KC >> 1) * WPAIR_STRIDE) * 4u;

    const int t    = threadIdx.x;
    const int lane = t & 31;
    const int w    = t >> 5;
    const int cg   = blockIdx.y;               // column group of 128
    const int sub  = lane & 15;
    const int half = lane >> 4;
    const int colL = (w << 4) + sub;           // 0..127 within group
    const int col  = (cg << 7) + colL;         // global column
    const int nRT  = R >> 4;
    const float bcol = bias ? bias[col] : 0.0f;

    float sAcc = 0.0f, qAcc = 0.0f;

    for (int rt = blockIdx.x; rt < nRT; rt += gridDim.x) {
        const int r0 = rt << 4;
        // prefetch next slab of A rows into L2 (global_prefetch_b8)
        {
            int rtn = rt + gridDim.x;
            if (rtn < nRT && t < 16)
                __builtin_prefetch(&A[(size_t)(((size_t)rtn << 4) + t) * lda], 0, 1);
        }
        v8f acc = {};
        for (int kc0 = 0; kc0 < Kd; kc0 += KC) {
            __syncthreads();
            if (MODE == 0) {
                // async global->LDS staging, pair-interleaved scatter
                for (int p = t; p < (KC >> 1) << 7; p += 256) {
                    int k2 = p >> 7;
                    int c  = p & 127;
                    const float* wp = W + (size_t)(kc0 + (k2 << 1)) * C + (cg << 7) + c;
                    unsigned l0 = ldsWs + (unsigned)(k2 * WPAIR_STRIDE + (c << 1)) * 4u;
                    async_ld_b32(l0, wp);
                    async_ld_b32(l0 + 4u, wp + C);
                }
                for (int p = t; p < (KC << 4); p += 256) {
                    int r = p / KC;
                    int k = p - r * KC;
                    const float* ap = &A[(size_t)(r0 + r) * lda + kc0 + k];
                    unsigned l = ldsAs + (unsigned)(((k >> 1) << 5) + (r << 1) + (k & 1)) * 4u;
                    async_ld_b32(l, ap);
                }
                wait_async0();
            } else {
                // synchronous staging with BN+GELU applied in registers
                for (int p = t; p < (KC >> 1) << 7; p += 256) {
                    int k2 = p >> 7;
                    int c  = p & 127;
                    const float* wp = W + (size_t)(kc0 + (k2 << 1)) * C + (cg << 7) + c;
                    v2f wv = { wp[0], wp[C] };
                    *(v2f*)&Ws[(size_t)k2 * WPAIR_STRIDE + (c << 1)] = wv;
                }
                for (int p = t; p < (KC << 4); p += 256) {
                    int r = p / KC;
                    int k = p - r * KC;
                    float x = A[(size_t)(r0 + r) * lda + kc0 + k];
                    int gk = kc0 + k;
                    x = gelu_f(fmaf(x, inS[gk], inB[gk]));
                    As[((k >> 1) << 5) + (r << 1) + (k & 1)] = x;
                }
            }
            __syncthreads();
            // pure LDS + WMMA inner loop
#pragma unroll 4
            for (int k = 0; k < KC; k += 4) {
                const int k2 = (k >> 1) + half;
                v2f av = *(const v2f*)&As[(k2 << 5) + (sub << 1)];
                v2f bv = *(const v2f*)&Ws[(size_t)k2 * WPAIR_STRIDE + (colL << 1)];
                acc = __builtin_amdgcn_wmma_f32_16x16x4_f32(
                    false, av, false, bv, (short)0, acc, false, false);
            }
        }
        float* Yp = Y + (size_t)(r0 + half * 8) * C + col;
#pragma unroll
        for (int i = 0; i < 8; ++i) {
            float v = acc[i] + bcol;
            Yp[(size_t)i * C] = v;
            sAcc += v;
            qAcc += v * v;
        }
    }
    if (partS) {
        sAcc += __shfl_down(sAcc, 16, 32);
        qAcc += __shfl_down(qAcc, 16, 32);
        if (lane < 16) {
            partS[(size_t)blockIdx.x * C + col] = sAcc;
            partQ[(size_t)blockIdx.x * C + col] = qAcc;
        }
    }
}

// Reduce per-block partials -> BN scale/shift (deterministic, fixed order).
__global__ void bn_finalize(const float* __restrict__ partS, const float* __restrict__ partQ,
                            int P, int C, float invR,
                            const float* __restrict__ g, const float* __restrict__ bt,
                            float* __restrict__ outS, float* __restrict__ outB) {
    int c = blockIdx.x * blockDim.x + threadIdx.x;
    if (c >= C) return;
    float s = 0.0f, q = 0.0f;
    for (int p = 0; p < P; ++p) {
        s += partS[(size_t)p * C + c];
        q += partQ[(size_t)p * C + c];
    }
    float m = s * invR;
    float v = q * invR - m * m;
    v = fmaxf(v, 0.0f);
    float sc = g[c] / sqrtf(v + 1e-5f);
    outS[c] = sc;
    outB[c] = bt[c] - m * sc;
}

// BN+GELU on y2, max-pool over K=32, add into new_feat (skip already there).
__global__ __launch_bounds__(256) void pool_kernel(const float* __restrict__ y2,
                                                   const float* __restrict__ s,
                                                   const float* __restrict__ b,
                                                   float* __restrict__ nf) {
    int i = blockIdx.x * blockDim.x + threadIdx.x;   // RPTS*COUT threads
    int c = i & (COUT - 1);
    size_t bm = (size_t)(i >> 7);
    float sc = s[c], sh = b[c];
    const float* p = y2 + bm * KK * COUT + c;
    float mx = -3.0e38f;
#pragma unroll
    for (int kk = 0; kk < KK; ++kk) {
        float v = p[(size_t)kk * COUT] * sc + sh;
        mx = fmaxf(mx, gelu_f(v));
    }
    nf[i] += mx;
}

// out = new_feat + BN(y4)   (InvResMLP residual)
__global__ __launch_bounds__(256) void final_kernel(const float* __restrict__ nf,
                                                    const float* __restrict__ y4,
                                                    const float* __restrict__ s,
                                                    const float* __restrict__ b,
                                                    float* __restrict__ out) {
    int i = blockIdx.x * blockDim.x + threadIdx.x;
    int c = i & (COUT - 1);
    out[i] = nf[i] + y4[i] * s[c] + b[c];
}

// ---------------------------------------------------------------------------
extern "C" void kernel_launch(void* const* d_in, const int* in_sizes, int n_in,
                              void* d_out, int out_size, void* d_ws, size_t ws_size,
                              hipStream_t stream) {
    const float* feat = (const float*)d_in[0];
    const float* pos  = (const float*)d_in[1];
    const float* lw1  = (const float*)d_in[2];
    const float* lb1  = (const float*)d_in[3];
    const float* lg1  = (const float*)d_in[4];
    const float* lbt1 = (const float*)d_in[5];
    const float* lw2  = (const float*)d_in[6];
    const float* lb2  = (const float*)d_in[7];
    const float* lg2  = (const float*)d_in[8];
    const float* lbt2 = (const float*)d_in[9];
    const float* skw  = (const float*)d_in[10];
    const float* bw1  = (const float*)d_in[11];
    const float* bb1  = (const float*)d_in[12];
    const float* bg1  = (const float*)d_in[13];
    const float* bbt1 = (const float*)d_in[14];
    const float* bw2  = (const float*)d_in[15];
    const float* bb2  = (const float*)d_in[16];
    const float* bg2  = (const float*)d_in[17];
    const float* bbt2 = (const float*)d_in[18];

    char* ws = (char*)d_ws;
    size_t cur = 0;
    auto take = [&](size_t bytes) -> char* {
        char* p = ws + cur;
        cur += (bytes + 255) & ~(size_t)255;
        return p;
    };

    float* sc1 = (float*)take(COUT * 4);
    float* sh1 = (float*)take(COUT * 4);
    float* sc2 = (float*)take(COUT * 4);
    float* sh2 = (float*)take(COUT * 4);
    float* sc3 = (float*)take(CMID * 4);
    float* sh3 = (float*)take(CMID * 4);
    float* sc4 = (float*)take(COUT * 4);
    float* sh4 = (float*)take(COUT * 4);
    float* partS = (float*)take((size_t)1024 * CMID * 4);   // max P*C
    float* partQ = (float*)take((size_t)1024 * CMID * 4);
    int*   cidx  = (int*)take((size_t)RPTS * 4);
    int*   knn   = (int*)take((size_t)RLOC * 4);
    float* cpos  = (float*)take((size_t)RPTS * 3 * 4);
    float* cfeat = (float*)take((size_t)RPTS * CIN * 4);
    float* lw1p  = (float*)take((size_t)LIN * COUT * 4);
    float* lin   = (float*)take((size_t)RLOC * LIN * 4);
    float* y1    = (float*)take((size_t)RLOC * COUT * 4);
    float* y2    = (float*)take((size_t)RLOC * COUT * 4);
    float* nfeat = (float*)take((size_t)RPTS * COUT * 4);
    float* y3    = (float*)take((size_t)RPTS * CMID * 4);
    float* y4    = (float*)take((size_t)RPTS * COUT * 4);

    float* out_feat = (float*)d_out;                       // (B,M,128)
    float* out_cpos = (float*)d_out + (size_t)RPTS * COUT; // (B,M,3)

    const float invRloc = 1.0f / (float)RLOC;
    const float invRpts = 1.0f / (float)RPTS;
    auto ldsb = [](int KC) -> size_t {
        return (size_t)((KC / 2) * WPAIR_STRIDE + KC * 16) * 4;
    };

    pad_lw1_kernel<<<(LIN * COUT + 255) / 256, 256, 0, stream>>>(lw1, lw1p);
    fps_kernel<<<BB, 1024, 0, stream>>>(pos, cidx);
    gather_centers<<<RPTS / 4, 256, 0, stream>>>(feat, pos, cidx, cpos, cfeat, out_cpos);
    ballquery_kernel<<<RPTS, 256, 0, stream>>>(pos, cpos, knn);
    gather_local<<<RLOC / 4, 256, 0, stream>>>(feat, pos, knn, cpos, lin);

    // y1 = local_in @ lw1 + lb1            (R=524288, C=128, Kd=KC=68) P=1024
    gemm_core<0><<<dim3(1024, 1), 256, ldsb(LIN), stream>>>(
        lin, LIN, lw1p, lb1, nullptr, nullptr, y1, partS, partQ, RLOC, COUT, LIN, LIN);
    bn_finalize<<<1, COUT, 0, stream>>>(partS, partQ, 1024, COUT, invRloc, lg1, lbt1, sc1, sh1);

    // y2 = gelu(bn(y1)) @ lw2 + lb2        (R=524288, C=128, Kd=KC=128) P=1024
    gemm_core<2><<<dim3(1024, 1), 256, ldsb(COUT), stream>>>(
        y1, COUT, lw2, lb2, sc1, sh1, y2, partS, partQ, RLOC, COUT, COUT, COUT);
    bn_finalize<<<1, COUT, 0, stream>>>(partS, partQ, 1024, COUT, invRloc, lg2, lbt2, sc2, sh2);

    // new_feat = center_feat @ skip_w      (R=16384, C=128, Kd=KC=64)
    gemm_core<0><<<dim3(256, 1), 256, ldsb(CIN), stream>>>(
        cfeat, CIN, skw, nullptr, nullptr, nullptr, nfeat, nullptr, nullptr, RPTS, COUT, CIN, CIN);
    // new_feat += maxpool_K(gelu(bn(y2)))
    pool_kernel<<<RPTS * COUT / 256, 256, 0, stream>>>(y2, sc2, sh2, nfeat);

    // y3 = new_feat @ bw1 + bb1            (R=16384, C=512, Kd=KC=128) P=256
    gemm_core<0><<<dim3(256, 4), 256, ldsb(COUT), stream>>>(
        nfeat, COUT, bw1, bb1, nullptr, nullptr, y3, partS, partQ, RPTS, CMID, COUT, COUT);
    bn_finalize<<<2, 256, 0, stream>>>(partS, partQ, 256, CMID, invRpts, bg1, bbt1, sc3, sh3);

    // y4 = gelu(bn(y3)) @ bw2 + bb2        (R=16384, C=128, Kd=512, KC=128) P=256
    gemm_core<2><<<dim3(256, 1), 256, ldsb(COUT), stream>>>(
        y3, CMID, bw2, bb2, sc3, sh3, y4, partS, partQ, RPTS, COUT, CMID, COUT);
    bn_finalize<<<1, COUT, 0, stream>>>(partS, partQ, 256, COUT, invRpts, bg2, bbt2, sc4, sh4);

    // out = new_feat + bn(y4)
    final_kernel<<<RPTS * COUT / 256, 256, 0, stream>>>(nfeat, y4, sc4, sh4, out_feat);
}